// Build_LPyr_46231027974207
// MI455X (gfx1250) — compile-verified
//
// MI455X (gfx1250) Laplacian pyramid, 5 levels, NCHW f32 (16,3,1024,1024).
// Bandwidth-bound (~0.5 GB total traffic @ 23.3 TB/s). Strategy:
//  - gpyr[l] is materialized directly into the output slot for level l,
//    then overwritten in place by the Laplacian (elementwise) -> no scratch.
//  - Input tiles staged into LDS with gfx1250 async DMA
//    (global_load_async_to_lds_b32 + s_wait_asynccnt).
//  - Expand is expressed as a banded matmul and executed with
//    V_WMMA_F32_16X16X4_F32 (3 WMMAs per 16x16 fine tile), full f32.
//  - All fragment construction is branch-free (v_cndmask only) so EXEC stays
//    all-ones through the WMMA region and no divergent wait chains appear.
#include <hip/hip_runtime.h>
#include <stdint.h>

typedef __attribute__((ext_vector_type(2))) float v2f;
typedef __attribute__((ext_vector_type(8))) float v8f;

__device__ __forceinline__ int clampi(int v, int lo, int hi) {
    return v < lo ? lo : (v > hi ? hi : v);
}

// gfx1250 async global->LDS copy (GV mode, saddr=off). VDST = LDS byte
// address VGPR, VADDR = 64-bit global address pair. Tracked by ASYNCcnt.
__device__ __forceinline__ void async_ld_f32(uint32_t lds_off, const float* g) {
    asm volatile("global_load_async_to_lds_b32 %0, %1, off"
                 :: "v"(lds_off), "v"((unsigned long long)(uintptr_t)g)
                 : "memory");
}
__device__ __forceinline__ void wait_async0() {
    asm volatile("s_wait_asynccnt 0x0" ::: "memory");
}

// ---------------------------------------------------------------------------
// Reduce: out[y][x] = sum_{r,c in 0..4} h[r]h[c] * in[clamp(2y+r-2)][clamp(2x+c-2)]
// Block: 256 threads -> 8x64 output tile per plane. Input tile 19x131 in LDS.
// ---------------------------------------------------------------------------
__global__ __launch_bounds__(256) void pyr_reduce_kernel(
        const float* __restrict__ src, float* __restrict__ dst, int H, int W) {
    const int Ho = H >> 1, Wo = W >> 1;
    __shared__ float sIn[19 * 132];
    const int t = (int)threadIdx.x;
    const int plane = (int)blockIdx.z;
    const int y0b = (int)blockIdx.y * 8;
    const int x0b = (int)blockIdx.x * 64;
    const float* sp = src + (size_t)plane * H * W;
    float* dp = dst + (size_t)plane * Ho * Wo;

    const uint32_t sbase = (uint32_t)(uintptr_t)sIn;  // low 32b of flat = LDS addr
    const int TOT = 19 * 132;                          // 2508 elems, 10 iters x 256
    #pragma unroll
    for (int i = 0; i < 10; ++i) {
        int idx = t + i * 256;
        idx = idx < TOT ? idx : TOT - 1;               // avoid divergence in tail
        int lr = idx / 132;
        int lc = idx - lr * 132;
        int gy = clampi(2 * y0b - 2 + lr, 0, H - 1);   // edge clamp folded here
        int gx = clampi(2 * x0b - 2 + lc, 0, W - 1);
        async_ld_f32(sbase + 4u * (uint32_t)idx, sp + (size_t)gy * W + gx);
    }
    wait_async0();
    __syncthreads();

    const float hv[5] = {0.0625f, 0.25f, 0.375f, 0.25f, 0.0625f};
    #pragma unroll
    for (int o = 0; o < 2; ++o) {                      // 512 outputs / 256 threads
        int lin = t + o * 256;
        int oy = lin >> 6;
        int ox = lin & 63;
        const float* row = &sIn[(2 * oy) * 132 + 2 * ox];
        float acc = 0.f;
        #pragma unroll
        for (int r = 0; r < 5; ++r) {
            const float* rr = row + r * 132;
            float s = hv[0] * rr[0] + hv[1] * rr[1] + hv[2] * rr[2] +
                      hv[3] * rr[3] + hv[4] * rr[4];
            acc += hv[r] * s;
        }
        dp[(size_t)(y0b + oy) * Wo + (x0b + ox)] = acc;
    }
}

// ---------------------------------------------------------------------------
// Expand + subtract via WMMA f32 16x16x4.
// Fine tile per wave: 16 rows x 16 cols at (Y0b, X0b + 16*wave).
//   A(16x12): A[m][k] = vertical-expand of coarse col (X0/2 - 1 + k) at fine
//             row Y0b+m (even rows: 0.125/0.75/0.125; odd rows: 0.5/0.5),
//             image clamping folded into the LDS tile load.
//   B(12x16): banded horizontal taps: even n -> {n/2:0.125, n/2+1:0.75,
//             n/2+2:0.125}; odd n -> {(n+1)/2:0.5, (n+1)/2+1:0.5}.
//   D = A x B  == expand(g1) on the tile;  out = g0 - D  (in-place safe).
// Fragment layouts (f32, wave32):
//   A 16x4: lane<16 -> (M=lane, K=kb), lane>=16 -> (M=lane-16, K=kb+2);
//           VGPR e holds K=kb+e.  B 4x16: lane<16 -> (K=kb, N=lane),
//           lane>=16 -> (K=kb+2, N=lane-16); VGPR e holds K=kb+e.
//   D: VGPR j -> (M = j + 8*(lane>=16), N = lane&15).
// ---------------------------------------------------------------------------
__global__ __launch_bounds__(256) void pyr_expand_sub_kernel(
        const float* __restrict__ g0, const float* __restrict__ g1,
        float* __restrict__ dst, int H1, int W1) {
    const int Wf = W1 * 2;
    __shared__ float sG[10 * 68];                      // coarse tile 10 rows x 68 cols
    const int t = (int)threadIdx.x;
    const int plane = (int)blockIdx.z;
    const int Y0b = (int)blockIdx.y * 16;              // fine rows
    const int X0b = (int)blockIdx.x * 128;             // fine cols (8 waves x 16)
    const float* g1p = g1 + (size_t)plane * H1 * W1;
    const float* g0p = g0 + (size_t)plane * (2 * H1) * Wf;
    float* dp = dst + (size_t)plane * (2 * H1) * Wf;

    const int ycb = (Y0b >> 1) - 1;                    // coarse row of LDS row 0
    const int xcb0 = (X0b >> 1) - 1;                   // coarse col of LDS col 0
    const uint32_t sbase = (uint32_t)(uintptr_t)sG;
    #pragma unroll
    for (int i = 0; i < 3; ++i) {                      // 680 elems, 3 iters x 256
        int idx = t + 256 * i;
        idx = idx < 680 ? idx : 679;
        int lr = idx / 68;
        int lc = idx - lr * 68;
        int gy = clampi(ycb + lr, 0, H1 - 1);
        int gx = clampi(xcb0 + lc, 0, W1 - 1);
        async_ld_f32(sbase + 4u * (uint32_t)idx, g1p + (size_t)gy * W1 + gx);
    }
    wait_async0();
    __syncthreads();

    const int lane  = t & 31;
    const int wv    = t >> 5;        // wave id 0..7 (wave32)
    const int mn    = lane & 15;     // M for A-frag, N for B-frag / D
    const int khalf = lane >> 4;     // K split across half-waves (+2 upper)
    const int lr0   = (mn >> 1) + 1; // LDS row of coarse row y0 for fine row Y0b+mn
    const int par   = mn & 1;        // row parity (A) == col parity (B)

    // Branch-free vertical weights: even -> (0.125, 0.75, 0.125); odd -> (0, .5, .5)
    const float wm = par ? 0.0f : 0.125f;
    const float w0 = par ? 0.5f : 0.75f;
    const float w1 = par ? 0.5f : 0.125f;
    // Horizontal band base: even n -> n/2, odd n -> (n+1)/2
    const int hbase = (mn + par) >> 1;

    v8f acc = {};
    #pragma unroll
    for (int kc = 0; kc < 3; ++kc) {                   // K = 12 in chunks of 4
        const int kb = 4 * kc + 2 * khalf;
        float a[2], b[2];
        #pragma unroll
        for (int e = 0; e < 2; ++e) {                  // VGPR e holds K = kb + e
            const int k  = kb + e;
            const int lc = 8 * wv + k;                 // <= 8*7 + 11 = 67
            // A: always read 3 rows (lr0-1 >= 0 always), weights select parity.
            float rm = sG[(lr0 - 1) * 68 + lc];
            float r0 = sG[lr0 * 68 + lc];
            float r1 = sG[(lr0 + 1) * 68 + lc];
            a[e] = wm * rm + w0 * r0 + w1 * r1;
            // B: banded weight, branch-free.
            const int d = k - hbase;
            float we = (d == 1) ? 0.75f
                     : ((d == 0) || (d == 2)) ? 0.125f : 0.0f;
            float wo = ((d == 0) || (d == 1)) ? 0.5f : 0.0f;
            b[e] = par ? wo : we;
        }
        v2f A = {a[0], a[1]};
        v2f B = {b[0], b[1]};
        acc = __builtin_amdgcn_wmma_f32_16x16x4_f32(
            /*neg_a=*/false, A, /*neg_b=*/false, B,
            /*c_mod=*/(short)0, acc, /*reuse_a=*/false, /*reuse_b=*/false);
    }

    // D layout: VGPR j -> (M = j + 8*khalf, N = mn). Fused subtract + store.
    const int X = X0b + 16 * wv + mn;
    #pragma unroll
    for (int j = 0; j < 8; ++j) {
        const int Y = Y0b + j + 8 * khalf;
        const size_t off = (size_t)Y * Wf + X;
        dp[off] = g0p[off] - acc[j];
    }
}

// ---------------------------------------------------------------------------
// Host launcher. Output layout: lpyr0 | lpyr1 | lpyr2 | lpyr3 | gpyr4, each
// NCHW-flat. Passes: (1) reduce chain writes gpyr[l] into slot l (slot 4 is
// final), (2) for l=0..3 overwrite slot l in place with gpyr[l]-expand(slot l+1).
// ---------------------------------------------------------------------------
extern "C" void kernel_launch(void* const* d_in, const int* in_sizes, int n_in,
                              void* d_out, int out_size, void* d_ws, size_t ws_size,
                              hipStream_t stream) {
    (void)in_sizes; (void)n_in; (void)out_size; (void)d_ws; (void)ws_size;
    const float* im = (const float*)d_in[0];
    float* out = (float*)d_out;
    const int NC = 16 * 3;
    const int dim[5] = {1024, 512, 256, 128, 64};
    size_t off[5];
    off[0] = 0;
    for (int l = 1; l < 5; ++l)
        off[l] = off[l - 1] + (size_t)NC * dim[l - 1] * dim[l - 1];

    // Pass 1: reduce chain (slot l+1 <- reduce(slot l)); slot 4 == lpyr4.
    for (int l = 0; l < 4; ++l) {
        const float* src = (l == 0) ? im : out + off[l];
        float* dst = out + off[l + 1];
        dim3 grid(dim[l + 1] / 64, dim[l + 1] / 8, NC);
        pyr_reduce_kernel<<<grid, dim3(256), 0, stream>>>(src, dst, dim[l], dim[l]);
    }
    // Pass 2: slot l <- gpyr[l] - expand(slot l+1), increasing l (slot l+1
    // still holds gpyr[l+1] when read).
    for (int l = 0; l < 4; ++l) {
        const float* g0 = (l == 0) ? im : out + off[l];
        const float* g1 = out + off[l + 1];
        float* dst = out + off[l];
        dim3 grid(dim[l] / 128, dim[l] / 16, NC);
        pyr_expand_sub_kernel<<<grid, dim3(256), 0, stream>>>(g0, g1, dst,
                                                              dim[l + 1], dim[l + 1]);
    }
}